// Covariance_26534307955220
// MI455X (gfx1250) — compile-verified
//
#include <hip/hip_runtime.h>

typedef __attribute__((ext_vector_type(2))) float v2f;
typedef __attribute__((ext_vector_type(4))) float v4f;
typedef __attribute__((ext_vector_type(8))) float v8f;

#define B_ 8
#define T_ 512
#define F_ 513
#define NPAIR 36

// np.triu_indices(8)
__device__ const unsigned char d_iu0[NPAIR] = {0,0,0,0,0,0,0,0, 1,1,1,1,1,1,1, 2,2,2,2,2,2, 3,3,3,3,3, 4,4,4,4, 5,5,5, 6,6, 7};
__device__ const unsigned char d_iu1[NPAIR] = {0,1,2,3,4,5,6,7, 1,2,3,4,5,6,7, 2,3,4,5,6,7, 3,4,5,6,7, 4,5,6,7, 5,6,7, 6,7, 7};

// One block (8 wave32) per (b,f).  G = sum_t c_t c_t^T  via V_WMMA_F32_16X16X4_F32,
// where c_t = 16 contiguous floats Xs[b,t,f,:,:].  A and B fragments are identical
// registers (A = C^T slab 16x4, B = C slab 4x16, mirrored lane layouts).
__global__ __launch_bounds__(256) void Covariance_26534307955220_kernel(
    const float* __restrict__ in, float* __restrict__ out) {
  const int bf   = blockIdx.x;          // 0 .. B_*F_-1
  const int b    = bf / F_;
  const int f    = bf - b * F_;
  const int tid  = threadIdx.x;
  const int wave = tid >> 5;
  const int lane = tid & 31;            // wave32
  const int m    = lane & 15;           // row (=col) index of this lane's A/B element
  const int q0   = (lane < 16) ? 0 : 2; // K offset held by this half-wave

  __shared__ float G[16 * 16];                       // accumulated Gram matrix
  __shared__ __align__(16) float outbuf[2 * NPAIR];  // 72 final values for this (b,f)

  G[tid] = 0.0f;                        // 256 threads zero 256 floats
  __syncthreads();

  // ---- Phase 1: WMMA reduction over T.  Each wave owns 64 t values. ----
  const size_t rowstride = (size_t)F_ * 16;                       // floats between t and t+1
  const float* base = in + ((size_t)b * T_ * F_ + f) * 16 + m;    // lane's column of C
  const int t0 = wave * 64;

  v8f acc = {};
  #pragma unroll
  for (int it = 0; it < 16; ++it) {
    const int t = t0 + it * 4 + q0;
    v2f a;
    a.x = __builtin_nontemporal_load(&base[(size_t)t * rowstride]);        // c_{t}  [m]
    a.y = __builtin_nontemporal_load(&base[(size_t)(t + 1) * rowstride]);  // c_{t+1}[m]
    // D = A*B + C ; A (16x4 f32) and B (4x16 f32) share the same register pair here.
    acc = __builtin_amdgcn_wmma_f32_16x16x4_f32(
        /*neg_a=*/false, a, /*neg_b=*/false, a,
        /*c_mod=*/(short)0, acc, /*reuse_a=*/false, /*reuse_b=*/false);
  }

  // ---- Combine 8 per-wave partial Gram matrices in LDS (ds_add_f32). ----
  // D layout: vgpr r, lanes 0-15 -> G[r, lane]; lanes 16-31 -> G[r+8, lane-16].
  {
    const int rowoff = (lane < 16) ? 0 : 8;
    #pragma unroll
    for (int r = 0; r < 8; ++r)
      atomicAdd(&G[(rowoff + r) * 16 + m], acc[r]);
  }
  __syncthreads();

  // ---- Phase 2: 72 outputs from G (layout: [re(36) | im(36)]). ----
  if (tid < 2 * NPAIR) {
    const int  c = (tid >= NPAIR);
    const int  p = c ? tid - NPAIR : tid;
    const int  i = d_iu0[p], j = d_iu1[p];
    float v;
    if (!c) v = G[i * 16 + j]       + G[(i + 8) * 16 + (j + 8)];  // re: Re^T Re + Im^T Im
    else    v = G[i * 16 + (j + 8)] - G[(i + 8) * 16 + j];        // im: Re^T Im - Im^T Re
    outbuf[tid] = v * (1.0f / (float)T_);                          // mean over T
  }
  __syncthreads();

  // ---- Phase 3: broadcast the mean to all t — 605 MB streaming write.      ----
  // 72 floats = 18 v4f per (b,t,f); record offset is 288 B => 16 B aligned.
  // Non-temporal B128 stores: write-once data must not displace L2 lines.
  const v4f* ob4 = (const v4f*)outbuf;
  v4f* obase4 = (v4f*)(out + ((size_t)b * T_ * F_ + f) * (2 * NPAIR));
  const size_t tstride4 = (size_t)F_ * 18;          // v4f's between t and t+1
  for (int idx = tid; idx < T_ * 18; idx += 256) {
    const int t = idx / 18;
    const int r = idx - t * 18;
    __builtin_nontemporal_store(ob4[r], &obase4[(size_t)t * tstride4 + r]);
  }
}

extern "C" void kernel_launch(void* const* d_in, const int* in_sizes, int n_in,
                              void* d_out, int out_size, void* d_ws, size_t ws_size,
                              hipStream_t stream) {
  (void)in_sizes; (void)n_in; (void)d_ws; (void)ws_size; (void)out_size;
  const float* Xs = (const float*)d_in[0];
  float* out = (float*)d_out;
  dim3 grid(B_ * F_);   // 4104 blocks, one per (b,f)
  dim3 block(256);      // 8 wave32
  Covariance_26534307955220_kernel<<<grid, block, 0, stream>>>(Xs, out);
}